// GraphSAGE_16707422781625
// MI455X (gfx1250) — compile-verified
//
#include <hip/hip_runtime.h>

// GraphSAGE 2-layer forward for MI455X (gfx1250, wave32).
// Layer kernel: gather+mean into LDS, then fp32 WMMA (V_WMMA_F32_16X16X4_F32)
// for x[32,256] @ W[256,128], bias folded into accumulator init, ReLU on store.

typedef __attribute__((ext_vector_type(2))) float v2f;
typedef __attribute__((ext_vector_type(8))) float v8f;

#define DIMC 128          // feature dim
#define DCNT 16           // neighbor fan-out
#define KTOT 256          // concat dim (self 128 + neigh-mean 128)
#define MT   32           // nodes per workgroup
#define ROWSTRIDE (KTOT + 4)  // 260 floats: bank-conflict-free A-fragment reads

__global__ __launch_bounds__(256)
void sage_layer_kernel(const float* __restrict__ h_prev,
                       const int*   __restrict__ self_idx,
                       const int*   __restrict__ neigh_idx,
                       const float* __restrict__ W,      // [256,128] row-major
                       const float* __restrict__ bias,   // [128]
                       float*       __restrict__ out,    // [n_nodes,128]
                       int n_nodes)
{
    __shared__ float xs[MT * ROWSTRIDE];

    const int tid  = threadIdx.x;
    const int lane = tid & 31;
    const int wave = tid >> 5;              // 8 waves
    const int base = blockIdx.x * MT;
    const bool full = (base + MT) <= n_nodes;   // wave-uniform fast path

    // Warm this wave's W column slice (cols wave*16 .. wave*16+15) into cache
    // while the gather phase is running.  Emits global_prefetch_b8.
    {
        const float* wp = W + (size_t)lane * 8 * DIMC + wave * 16;
        __builtin_prefetch(wp, 0, 3);
        __builtin_prefetch(wp + 4 * DIMC, 0, 3);
    }

    // ---------------- Phase 1: gather + mean into LDS -------------------
    // Each wave builds 4 node rows; lane handles float4 slice [lane*4, lane*4+4).
    for (int i = 0; i < 4; ++i) {
        const int ml    = wave * 4 + i;                 // local node 0..31
        const int node  = base + ml;
        const int nodec = full ? node : (node < n_nodes ? node : (n_nodes - 1));

        const int self = self_idx[nodec];
        const float4 sf =
            *(const float4*)(h_prev + (size_t)self * DIMC + lane * 4);
        *(float4*)(xs + ml * ROWSTRIDE + lane * 4) = sf;

        // lanes 0..15 fetch the 16 neighbor indices, broadcast via shfl
        const int nj = neigh_idx[(size_t)nodec * DCNT + (lane & 15)];
        float ax = 0.f, ay = 0.f, az = 0.f, aw = 0.f;
        #pragma unroll
        for (int j = 0; j < DCNT; ++j) {
            const int nidx = __shfl(nj, j, 32);
            const float4 nf =
                *(const float4*)(h_prev + (size_t)nidx * DIMC + lane * 4);
            ax += nf.x; ay += nf.y; az += nf.z; aw += nf.w;
        }
        const float s = 1.0f / (float)DCNT;
        float4 mn; mn.x = ax * s; mn.y = ay * s; mn.z = az * s; mn.w = aw * s;
        *(float4*)(xs + ml * ROWSTRIDE + DIMC + lane * 4) = mn;
    }
    __syncthreads();

    // ---------------- Phase 2: WMMA fp32 matmul -------------------------
    // Wave w owns output columns n0..n0+15 and both 16-row M tiles.
    const int n0    = wave * 16;
    const int col   = lane & 15;
    const int khalf = (lane >> 4) & 1;      // 0: K={k,k+1}, 1: K={k+2,k+3}

    const float bb = bias[n0 + col];        // same col for all 8 C VGPRs
    v8f c0, c1;
    #pragma unroll
    for (int r = 0; r < 8; ++r) { c0[r] = bb; c1[r] = bb; }

    const float* xrow0 = xs + (size_t)col * ROWSTRIDE;          // rows 0..15
    const float* xrow1 = xs + (size_t)(16 + col) * ROWSTRIDE;   // rows 16..31

    for (int k = 0; k < KTOT; k += 4) {
        const int kk = k + khalf * 2;
        // A fragments: one aligned b64 LDS load per lane, conflict-free
        // (compiler fuses consecutive k-steps into ds_load_2addr_b64).
        const v2f a0 = *(const v2f*)(xrow0 + kk);
        const v2f a1 = *(const v2f*)(xrow1 + kk);
        // B fragment: W[kk][n0+col], W[kk+1][n0+col] (L2-resident).
        v2f bv;
        bv.x = W[(size_t)kk * DIMC + n0 + col];
        bv.y = W[(size_t)(kk + 1) * DIMC + n0 + col];

        c0 = __builtin_amdgcn_wmma_f32_16x16x4_f32(
                 false, a0, false, bv, (short)0, c0, false, false);
        c1 = __builtin_amdgcn_wmma_f32_16x16x4_f32(
                 false, a1, false, bv, (short)0, c1, false, false);
    }

    // ---------------- Phase 3: ReLU + store ------------------------------
    const int rowoff = khalf * 8;           // C/D layout: half-wave row split
    float* outp = out + (size_t)base * DIMC + n0 + col;
    if (full) {
        // Common case: whole 32-node tile in range -> unguarded stores.
        #pragma unroll
        for (int r = 0; r < 8; ++r) {
            const int m0 = r + rowoff;      // c0 row (0..15)
            outp[(size_t)m0 * DIMC]        = fmaxf(c0[r], 0.0f);
            outp[(size_t)(16 + m0) * DIMC] = fmaxf(c1[r], 0.0f);
        }
    } else {
        #pragma unroll
        for (int r = 0; r < 8; ++r) {
            const int m0 = r + rowoff;
            const int m1 = 16 + m0;
            if (base + m0 < n_nodes) outp[(size_t)m0 * DIMC] = fmaxf(c0[r], 0.0f);
            if (base + m1 < n_nodes) outp[(size_t)m1 * DIMC] = fmaxf(c1[r], 0.0f);
        }
    }
}

extern "C" void kernel_launch(void* const* d_in, const int* in_sizes, int n_in,
                              void* d_out, int out_size, void* d_ws, size_t ws_size,
                              hipStream_t stream) {
    // setup_inputs() order:
    // 0 features [N0,128] f32   1 W1 [256,128] f32   2 b1 [128] f32
    // 3 W2 [256,128] f32        4 b2 [128] f32
    // 5 neigh1_idx [N1,16] i32  6 self1_idx [N1] i32
    // 7 neigh2_idx [N2,16] i32  8 self2_idx [N2] i32
    const float* features = (const float*)d_in[0];
    const float* W1       = (const float*)d_in[1];
    const float* b1       = (const float*)d_in[2];
    const float* W2       = (const float*)d_in[3];
    const float* b2       = (const float*)d_in[4];
    const int*   neigh1   = (const int*)d_in[5];
    const int*   self1    = (const int*)d_in[6];
    const int*   neigh2   = (const int*)d_in[7];
    const int*   self2    = (const int*)d_in[8];

    const int N1 = in_sizes[6];   // 100000
    const int N2 = in_sizes[8];   // 50000

    float* h1 = (float*)d_ws;     // needs N1*128*4 = 51.2 MB scratch

    const dim3 blk(256);
    sage_layer_kernel<<<dim3((N1 + MT - 1) / MT), blk, 0, stream>>>(
        features, self1, neigh1, W1, b1, h1, N1);
    sage_layer_kernel<<<dim3((N2 + MT - 1) / MT), blk, 0, stream>>>(
        h1, self2, neigh2, W2, b2, (float*)d_out, N2);
}